// SphericalAttention_17454747091586
// MI455X (gfx1250) — compile-verified
//
#include <hip/hip_runtime.h>
#include <hip/hip_bf16.h>
#include <math.h>

// ---------------- constants ----------------
#define NN     4096      // tokens (64*64)
#define CC     512       // channels
#define NHEAD  8
#define DH     64
#define C3     1536
#define NPAIR  16777216  // NN*NN

typedef __attribute__((ext_vector_type(2))) float v2f;
typedef __attribute__((ext_vector_type(8))) float v8f;

__device__ __forceinline__ v2f ld2(const float* p) {
    return *(const v2f*)p;
}

// D = A(16x4) * B(4x16) + C, fp32 WMMA (the only fp32 matrix op on CDNA5)
__device__ __forceinline__ v8f wmma4(v2f a, v2f b, v8f c) {
    return __builtin_amdgcn_wmma_f32_16x16x4_f32(
        /*neg_a=*/false, a, /*neg_b=*/false, b,
        /*c_mod=*/(short)0, c, /*reuse_a=*/false, /*reuse_b=*/false);
}

// ---------------- kernel 0: zero stats ----------------
__global__ void SA_zero_stats(double* stats) {
    stats[0] = 0.0;   // sum
    stats[1] = 0.0;   // sumsq
}

// ---------------- kernel 1: QKV GEMM (x @ w_qkv^T + b) -> head-major Q/K/V ----------------
// grid (C3/16=96, NN/16=256), block 32 (one wave per 16x16 tile)
__global__ void SA_qkv_gemm(const float* __restrict__ x, const float* __restrict__ w,
                            const float* __restrict__ bqkv,
                            float* __restrict__ qb, float* __restrict__ kb, float* __restrict__ vb) {
    const int lane = threadIdx.x;
    const int lm   = lane & 15;
    const int half = lane >> 4;
    const int tn   = blockIdx.x;   // tile over 1536 output cols
    const int tm   = blockIdx.y;   // tile over 4096 rows

    const float* arow = x + (size_t)(tm * 16 + lm) * CC;  // A row for this lane
    const float* brow = w + (size_t)(tn * 16 + lm) * CC;  // B col (= w row) for this lane

    v8f acc = {};
#pragma unroll 4
    for (int k0 = 0; k0 < CC; k0 += 4) {
        v2f af = ld2(&arow[k0 + half * 2]);
        v2f bf = ld2(&brow[k0 + half * 2]);
        acc = wmma4(af, bf, acc);
    }

    const int j  = tn * 16 + lm;          // global output col in [0,1536)
    const float bv = bqkv[j];
    const int which = j >> 9;             // 0=q, 1=k, 2=v
    const int hh    = (j >> 6) & 7;       // head
    const int d     = j & 63;             // dim in head
    float* dst = (which == 0) ? qb : ((which == 1) ? kb : vb);
#pragma unroll
    for (int r = 0; r < 8; ++r) {
        const int m = tm * 16 + r + half * 8;   // token index
        dst[((size_t)hh * NN + m) * DH + d] = acc[r] + bv;
    }
}

// ---------------- haversine ----------------
__device__ __forceinline__ float hav_dist(float lat_n, float lon_n, float lat_m, float lon_m) {
    const float sdlat = sinf(0.5f * (lat_m - lat_n));
    const float sdlon = sinf(0.5f * (lon_m - lon_n));
    float a = sdlat * sdlat + cosf(lat_n) * cosf(lat_m) * sdlon * sdlon;
    a = fminf(fmaxf(a, 0.0f), 1.0f);
    return 2.0f * asinf(sqrtf(a));
}

// ---------------- kernel 2: distance sum / sumsq (double accumulation) ----------------
__global__ void SA_dist_stats(const float* __restrict__ lat, const float* __restrict__ lon,
                              double* __restrict__ stats) {
    __shared__ double ssum[256];
    __shared__ double ssq[256];
    const int tid = threadIdx.x;
    double s = 0.0, q = 0.0;
    const int stride = gridDim.x * blockDim.x;
    for (int idx = blockIdx.x * blockDim.x + tid; idx < NPAIR; idx += stride) {
        const int n = idx >> 12;
        const int m = idx & (NN - 1);
        const float d = hav_dist(lat[n], lon[n], lat[m], lon[m]);
        s += (double)d;
        q += (double)d * (double)d;
    }
    ssum[tid] = s; ssq[tid] = q;
    __syncthreads();
    for (int off = 128; off > 0; off >>= 1) {
        if (tid < off) { ssum[tid] += ssum[tid + off]; ssq[tid] += ssq[tid + off]; }
        __syncthreads();
    }
    if (tid == 0) {
        atomicAdd(&stats[0], ssum[0]);
        atomicAdd(&stats[1], ssq[0]);
    }
}

// ---------------- kernel 3: bias[n][m] = -dist * 1/std (ddof=1) ----------------
__global__ void SA_bias_fill(const float* __restrict__ lat, const float* __restrict__ lon,
                             const double* __restrict__ stats, float* __restrict__ bias) {
    const double sum = stats[0];
    const double sq  = stats[1];
    const double cnt = (double)NPAIR;
    const double var = (sq - sum * sum / cnt) / (cnt - 1.0);
    const float neg_inv_std = (float)(-1.0 / sqrt(var));
    const int stride = gridDim.x * blockDim.x;
    for (int idx = blockIdx.x * blockDim.x + threadIdx.x; idx < NPAIR; idx += stride) {
        const int n = idx >> 12;
        const int m = idx & (NN - 1);
        bias[idx] = hav_dist(lat[n], lon[n], lat[m], lon[m]) * neg_inv_std;
    }
}

// ---------------- kernel 4: flash attention, one wave per (head, 16-query tile) ----------------
// grid (NN/16=256, NHEAD), block 32
__global__ void SA_attention(const float* __restrict__ Qb, const float* __restrict__ Kb,
                             const float* __restrict__ Vb, const float* __restrict__ biasb,
                             float* __restrict__ ao) {
    const int lane = threadIdx.x;
    const int lm   = lane & 15;
    const int half = lane >> 4;
    const int h    = blockIdx.y;
    const int qt   = blockIdx.x;
    const float scale = 0.125f;   // 1/sqrt(64)

    const float* Q = Qb + (size_t)h * NN * DH;
    const float* K = Kb + (size_t)h * NN * DH;
    const float* V = Vb + (size_t)h * NN * DH;

    __shared__ float pT[16 * 18];   // P tile, stride 18 -> 8B-aligned v2f LDS loads

    // preload Q tile A-fragments: a[j] = Q[qt*16+lm][kk*4 + 2*half + j]
    v2f qa[16];
#pragma unroll
    for (int kk = 0; kk < 16; ++kk)
        qa[kk] = ld2(&Q[(size_t)(qt * 16 + lm) * DH + kk * 4 + half * 2]);

    v8f acc[4] = {};               // out accumulator: 16 rows x 64 cols
    float mrow[8], lrow[8];
#pragma unroll
    for (int r = 0; r < 8; ++r) { mrow[r] = -__builtin_inff(); lrow[r] = 0.0f; }

    for (int kt = 0; kt < NN / 16; ++kt) {
        // ---- S = Q * K^T (16x16) ----
        v8f s = {};
#pragma unroll
        for (int kk = 0; kk < 16; ++kk) {
            v2f kf = ld2(&K[(size_t)(kt * 16 + lm) * DH + kk * 4 + half * 2]);
            s = wmma4(qa[kk], kf, s);
        }

        // ---- scale + bias, row max (reduce across lane%16 groups) ----
        float tmax[8];
#pragma unroll
        for (int r = 0; r < 8; ++r) {
            const int qrow = qt * 16 + r + half * 8;
            float sv = s[r] * scale + biasb[(size_t)qrow * NN + kt * 16 + lm];
            s[r] = sv;
            sv = fmaxf(sv, __shfl_xor(sv, 1, 16));
            sv = fmaxf(sv, __shfl_xor(sv, 2, 16));
            sv = fmaxf(sv, __shfl_xor(sv, 4, 16));
            sv = fmaxf(sv, __shfl_xor(sv, 8, 16));
            tmax[r] = sv;
        }

        // ---- online softmax update ----
#pragma unroll
        for (int r = 0; r < 8; ++r) {
            const float mnew = fmaxf(mrow[r], tmax[r]);
            const float corr = __expf(mrow[r] - mnew);   // 0 on first tile
            const float p    = __expf(s[r] - mnew);
            float ps = p;
            ps += __shfl_xor(ps, 1, 16);
            ps += __shfl_xor(ps, 2, 16);
            ps += __shfl_xor(ps, 4, 16);
            ps += __shfl_xor(ps, 8, 16);
            lrow[r] = lrow[r] * corr + ps;
            mrow[r] = mnew;
            s[r] = p;
            acc[0][r] *= corr; acc[1][r] *= corr; acc[2][r] *= corr; acc[3][r] *= corr;
        }

        // ---- transpose P through LDS (C/D layout -> A layout) ----
        __syncthreads();
#pragma unroll
        for (int r = 0; r < 8; ++r)
            pT[(r + half * 8) * 18 + lm] = s[r];
        __syncthreads();

        // ---- O += P(16x16) * V(16x64) ----
#pragma unroll
        for (int kk = 0; kk < 4; ++kk) {
            v2f pf = ld2(&pT[lm * 18 + kk * 4 + half * 2]);
            const int kidx = kt * 16 + kk * 4 + half * 2;
#pragma unroll
            for (int ct = 0; ct < 4; ++ct) {
                v2f vf;
                vf.x = V[(size_t)kidx * DH + ct * 16 + lm];
                vf.y = V[(size_t)(kidx + 1) * DH + ct * 16 + lm];
                acc[ct] = wmma4(pf, vf, acc[ct]);
            }
        }
    }

    // ---- normalize and store (head-major scratch layout) ----
#pragma unroll
    for (int ct = 0; ct < 4; ++ct) {
#pragma unroll
        for (int r = 0; r < 8; ++r) {
            const int n = qt * 16 + r + half * 8;
            ao[((size_t)h * NN + n) * DH + ct * 16 + lm] = acc[ct][r] / lrow[r];
        }
    }
}

// ---------------- kernel 5: output projection (ao @ w_out^T + b_out) ----------------
// grid (CC/16=32, NN/16=256), block 32
__global__ void SA_out_proj(const float* __restrict__ ao, const float* __restrict__ w,
                            const float* __restrict__ bo, float* __restrict__ out) {
    const int lane = threadIdx.x;
    const int lm   = lane & 15;
    const int half = lane >> 4;
    const int tn   = blockIdx.x;
    const int tm   = blockIdx.y;

    const int row = tm * 16 + lm;
    const float* brow = w + (size_t)(tn * 16 + lm) * CC;

    v8f acc = {};
#pragma unroll 4
    for (int k0 = 0; k0 < CC; k0 += 4) {
        const int c = k0 + half * 2;                 // flat channel = head*64 + d
        // A[row][c] lives at ao[(head*NN + row)*64 + d]; (c&63) even -> contiguous v2f
        v2f af = ld2(&ao[(((size_t)(c >> 6)) * NN + row) * DH + (c & 63)]);
        v2f bf = ld2(&brow[c]);
        acc = wmma4(af, bf, acc);
    }

    const int j  = tn * 16 + lm;
    const float bv = bo[j];
#pragma unroll
    for (int r = 0; r < 8; ++r) {
        const int m = tm * 16 + r + half * 8;
        out[(size_t)m * CC + j] = acc[r] + bv;
    }
}

// ---------------- launch ----------------
extern "C" void kernel_launch(void* const* d_in, const int* in_sizes, int n_in,
                              void* d_out, int out_size, void* d_ws, size_t ws_size,
                              hipStream_t stream) {
    const float* x     = (const float*)d_in[0];
    const float* lat   = (const float*)d_in[1];
    const float* lon   = (const float*)d_in[2];
    const float* w_qkv = (const float*)d_in[3];
    const float* b_qkv = (const float*)d_in[4];
    const float* w_out = (const float*)d_in[5];
    const float* b_out = (const float*)d_in[6];

    // workspace layout (floats): q | k | v | ao | bias | stats(double2)
    float* ws   = (float*)d_ws;
    float* qb   = ws;                          // 2,097,152 floats
    float* kb   = qb + (size_t)NN * CC;
    float* vb   = kb + (size_t)NN * CC;
    float* ao   = vb + (size_t)NN * CC;
    float* bias = ao + (size_t)NN * CC;        // 16,777,216 floats (67MB; L2-resident)
    double* stats = (double*)(bias + (size_t)NPAIR);   // byte offset divisible by 8

    SA_zero_stats<<<1, 1, 0, stream>>>(stats);
    SA_qkv_gemm<<<dim3(C3 / 16, NN / 16), 32, 0, stream>>>(x, w_qkv, b_qkv, qb, kb, vb);
    SA_dist_stats<<<4096, 256, 0, stream>>>(lat, lon, stats);
    SA_bias_fill<<<8192, 256, 0, stream>>>(lat, lon, stats, bias);
    SA_attention<<<dim3(NN / 16, NHEAD), 32, 0, stream>>>(qb, kb, vb, bias, ao);
    SA_out_proj<<<dim3(CC / 16, NN / 16), 32, 0, stream>>>(ao, w_out, b_out, (float*)d_out);
}